// SeparateWeightTensorProduct_24644522344489
// MI455X (gfx1250) — compile-verified
//
#include <hip/hip_runtime.h>

typedef __attribute__((ext_vector_type(2))) float v2f;
typedef __attribute__((ext_vector_type(8))) float v8f;

// LDS weight arena offsets (floats)
constexpr int O_W11 = 0;      // 64x64
constexpr int O_W12 = 4096;   // 64x32
constexpr int O_W13 = 6144;   // 32x32
constexpr int O_W14 = 7168;   // 32x64
constexpr int O_W15 = 9216;   // 32x16
constexpr int O_w21 = 9728;   // 64
constexpr int O_w22 = 9792;   // 32
constexpr int O_w23 = 9824;   // 32
constexpr int O_w24 = 9856;   // 64
constexpr int O_w25 = 9920;   // 16
constexpr int SW_TOT = 9936;

// epilogue constants
__device__ __constant__ float C000 = 0.10206207261596575f;  // sqrt(1/96)
__device__ __constant__ float C110 = 0.05892556509887896f;  // sqrt(1/96)/sqrt(3)
__device__ __constant__ float C011 = 0.10206207261596575f;  // sqrt(3/96)/sqrt(3)
__device__ __constant__ float C101 = 0.10206207261596575f;  // sqrt(3/96)/sqrt(3)
__device__ __constant__ float C112 = 0.125f;                // sqrt(3/32)/sqrt(6)

// One 16x16 output tile: D = A(16xK) * B(K x 16) using V_WMMA_F32_16X16X4_F32.
// A-frags preloaded in registers; B streamed from LDS weights (row-major, ldb).
template <int KSTEPS>
__device__ __forceinline__ v8f gemm_tile(const v2f* a, const float* bw, int ldb,
                                         int c0, int lane) {
  const int col = c0 + (lane & 15);
  const int h = lane >> 4;
  v8f acc = {};
#pragma unroll
  for (int k = 0; k < KSTEPS; ++k) {
    const int kk = 4 * k + 2 * h;
    v2f b;
    b.x = bw[kk * ldb + col];
    b.y = bw[(kk + 1) * ldb + col];
    acc = __builtin_amdgcn_wmma_f32_16x16x4_f32(false, a[k], false, b,
                                                (short)0, acc, false, false);
  }
  return acc;
}

__device__ __forceinline__ void copy_w(float* dst, const float* src, int cnt,
                                       int tid) {
  for (int i = tid; i < cnt; i += 256) dst[i] = src[i];
}

__global__ __launch_bounds__(256) void tp_sep_weight_kernel(
    const float* __restrict__ x1, const float* __restrict__ x2,
    const float* __restrict__ w11, const float* __restrict__ w21,
    const float* __restrict__ w12, const float* __restrict__ w22,
    const float* __restrict__ w13, const float* __restrict__ w23,
    const float* __restrict__ w14, const float* __restrict__ w24,
    const float* __restrict__ w15, const float* __restrict__ w25,
    float* __restrict__ out, int nAll) {
  __shared__ float sW[SW_TOT];
  __shared__ float sx2[8][16][4];

  const int tid = threadIdx.x;
  copy_w(sW + O_W11, w11, 4096, tid);
  copy_w(sW + O_W12, w12, 2048, tid);
  copy_w(sW + O_W13, w13, 1024, tid);
  copy_w(sW + O_W14, w14, 2048, tid);
  copy_w(sW + O_W15, w15, 512, tid);
  copy_w(sW + O_w21, w21, 64, tid);
  copy_w(sW + O_w22, w22, 32, tid);
  copy_w(sW + O_w23, w23, 32, tid);
  copy_w(sW + O_w24, w24, 64, tid);
  copy_w(sW + O_w25, w25, 16, tid);

  const int wave = tid >> 5;
  const int lane = tid & 31;
  const int s0 = (blockIdx.x * 8 + wave) * 16;  // sample base of this wave's tile

  if (s0 < nAll && lane < 16) {
    const int r = (s0 + lane < nAll) ? (s0 + lane) : (nAll - 1);
    const float* p = x2 + (size_t)r * 4;
    sx2[wave][lane][0] = p[0];
    sx2[wave][lane][1] = p[1];
    sx2[wave][lane][2] = p[2];
    sx2[wave][lane][3] = p[3];
  }
  __syncthreads();
  if (s0 >= nAll) return;

  const int nidx = lane & 15;
  const int h = lane >> 4;
  const int mrow = (s0 + nidx < nAll) ? (s0 + nidx) : (nAll - 1);  // clamp: keeps EXEC full
  const float* x1r = x1 + (size_t)mrow * 160;

  // ---- A fragments, kept live across all N-tiles of their K-group ----
  v2f a0[16];  // x1_0 : 16 samples x K=64
#pragma unroll
  for (int k = 0; k < 16; ++k) {
    const int c = 4 * k + 2 * h;
    a0[k].x = x1r[c];
    a0[k].y = x1r[c + 1];
  }
  v2f av[3][8];  // x1_1[:,:,i] : 16 samples x K=32, per component i
#pragma unroll
  for (int i = 0; i < 3; ++i) {
#pragma unroll
    for (int k = 0; k < 8; ++k) {
      const int u = 4 * k + 2 * h;
      av[i][k].x = x1r[64 + u * 3 + i];
      av[i][k].y = x1r[64 + (u + 1) * 3 + i];
    }
  }

  // ================= out0 : 64 cols = 4 tiles =================
  for (int t = 0; t < 4; ++t) {
    v8f acc = gemm_tile<16>(a0, sW + O_W11, 64, 16 * t, lane);  // A1
    const float w21v = sW[O_w21 + 16 * t + nidx];
    const float w24v = sW[O_w24 + 16 * t + nidx];
    float o[8];
#pragma unroll
    for (int r = 0; r < 8; ++r) {
      const float s2 = sx2[wave][r + 8 * h][0];
      o[r] = C000 * acc[r] * s2 * w21v;
    }
    for (int i = 0; i < 3; ++i) {
      v8f b = gemm_tile<8>(av[i], sW + O_W14, 64, 16 * t, lane);  // B1_i
#pragma unroll
      for (int r = 0; r < 8; ++r) {
        const float vi = sx2[wave][r + 8 * h][1 + i];
        o[r] += C110 * b[r] * vi * w24v;
      }
    }
#pragma unroll
    for (int r = 0; r < 8; ++r) {
      const int row = r + 8 * h;
      if (s0 + row < nAll)
        out[(size_t)(s0 + row) * 208 + 16 * t + nidx] = o[r];
    }
  }

  // ================= out1 : 32 cols x 3 = 2 tiles =================
  for (int tt = 0; tt < 2; ++tt) {
    v8f c1 = gemm_tile<16>(a0, sW + O_W12, 32, 16 * tt, lane);  // c1
    const float w22v = sW[O_w22 + 16 * tt + nidx];
    const float w23v = sW[O_w23 + 16 * tt + nidx];
    for (int i = 0; i < 3; ++i) {
      v8f d = gemm_tile<8>(av[i], sW + O_W13, 32, 16 * tt, lane);  // d1_i
#pragma unroll
      for (int r = 0; r < 8; ++r) {
        const int row = r + 8 * h;
        const float s2 = sx2[wave][row][0];
        const float vi = sx2[wave][row][1 + i];
        const float val = C011 * c1[r] * vi * w22v + C101 * d[r] * s2 * w23v;
        if (s0 + row < nAll)
          out[(size_t)(s0 + row) * 208 + 64 + (16 * tt + nidx) * 3 + i] = val;
      }
    }
  }

  // ================= out2 : 16 cols x 3 = 1 tile (cross product) =================
  float o2[3][8];
#pragma unroll
  for (int k = 0; k < 3; ++k)
#pragma unroll
    for (int r = 0; r < 8; ++r) o2[k][r] = 0.0f;

  for (int i = 0; i < 3; ++i) {
    v8f e = gemm_tile<8>(av[i], sW + O_W15, 16, 0, lane);  // e1_i
    const int j1 = (i + 1) % 3;
    const int j2 = (i + 2) % 3;
#pragma unroll
    for (int r = 0; r < 8; ++r) {
      const int row = r + 8 * h;
      const float vj1 = sx2[wave][row][1 + j1];
      const float vj2 = sx2[wave][row][1 + j2];
      o2[j2][r] += e[r] * vj1;  // eps_{i,i+1,i+2} = +1
      o2[j1][r] -= e[r] * vj2;  // eps_{i,i+2,i+1} = -1
    }
  }
  const float w25v = sW[O_w25 + nidx];
#pragma unroll
  for (int k = 0; k < 3; ++k) {
#pragma unroll
    for (int r = 0; r < 8; ++r) {
      const int row = r + 8 * h;
      if (s0 + row < nAll)
        out[(size_t)(s0 + row) * 208 + 160 + nidx * 3 + k] =
            C112 * w25v * o2[k][r];
    }
  }
}

extern "C" void kernel_launch(void* const* d_in, const int* in_sizes, int n_in,
                              void* d_out, int out_size, void* d_ws,
                              size_t ws_size, hipStream_t stream) {
  (void)n_in; (void)out_size; (void)d_ws; (void)ws_size;
  const float* x1 = (const float*)d_in[0];
  const float* x2 = (const float*)d_in[1];
  const float* w11 = (const float*)d_in[2];
  const float* w21 = (const float*)d_in[3];
  const float* w12 = (const float*)d_in[4];
  const float* w22 = (const float*)d_in[5];
  const float* w13 = (const float*)d_in[6];
  const float* w23 = (const float*)d_in[7];
  const float* w14 = (const float*)d_in[8];
  const float* w24 = (const float*)d_in[9];
  const float* w15 = (const float*)d_in[10];
  const float* w25 = (const float*)d_in[11];
  float* out = (float*)d_out;

  const int n = in_sizes[0] / 160;
  const int nTiles = (n + 15) / 16;   // 16 samples per wave
  const int blocks = (nTiles + 7) / 8;  // 8 waves (256 threads) per block

  tp_sep_weight_kernel<<<blocks, 256, 0, stream>>>(
      x1, x2, w11, w21, w12, w22, w13, w23, w14, w24, w15, w25, out, n);
}